// hypercorre_topk2_38276748542266
// MI455X (gfx1250) — compile-verified
//
#include <hip/hip_runtime.h>
#include <math.h>

// ---------------------------------------------------------------------------
// HyperCorre top-k forward for MI455X (gfx1250, wave32, WMMA f32 16x16x4).
//
// Input flattening assumption (jax sorted-dict tree order):
//   0:x0 1:r0 2:x1 3:r1 4:x2 5:r2 6:x3 7:r3
//   8..27  : hpn  -> l2{b,bt,g,w} l3{..} l32{..} l4{..} l43{..}
//   28..37 : s0   -> Wk,Wq,Wv,bk,bq,bv,dwB,dwW,ln_b,ln_g
//   38..47 : s1,  48..57 : s2,  58..67 : s3
// ---------------------------------------------------------------------------

typedef __attribute__((ext_vector_type(2))) float v2f;
typedef __attribute__((ext_vector_type(8))) float v8f;

#define S0TOT 1465
#define NQ0   4096
#define C0    64    // scale-0 channel count (pv/imp fast paths)

__device__ __forceinline__ int imin(int a, int b) { return a < b ? a : b; }

// ---------------------------------------------------------------------------
// Batched transpose: in [R, Cc] -> out [Cc, R]
// ---------------------------------------------------------------------------
__global__ void transpose_kernel(const float* __restrict__ in, float* __restrict__ out,
                                 int R, int Cc, long ibatch, long obatch) {
  int i = blockIdx.x * blockDim.x + threadIdx.x;
  if (i >= R * Cc) return;
  int r = i / Cc, c = i % Cc;
  out[(long)blockIdx.y * obatch + (long)c * R + r] =
      in[(long)blockIdx.y * ibatch + i];
}

// ---------------------------------------------------------------------------
// Wave-tile GEMM:  D[b] = A[b] (MxK row-major) * Bt^T (Bt is NxK row-major)
//                  + bias.  One wave = one 16x16 tile, v_wmma_f32_16x16x4_f32.
// Out-of-range rows/cols are CLAMPED (not guarded): their results are garbage
// but never stored, so the inner loop has no divergence -- one v2f load per
// operand per K-step.
// ---------------------------------------------------------------------------
__global__ void gemm_wmma_kernel(const float* __restrict__ A, long aB,
                                 const float* __restrict__ Bt,
                                 const float* __restrict__ bias,
                                 float* __restrict__ D, long dB,
                                 int M, int N, int K) {
  int wave = threadIdx.x >> 5;
  int lane = threadIdx.x & 31;
  int ntn = (N + 15) >> 4;
  int ntm = (M + 15) >> 4;
  int tile = blockIdx.x * (blockDim.x >> 5) + wave;
  if (tile >= ntm * ntn) return;                       // wave-uniform
  int tm = tile / ntn, tn = tile % ntn;
  const float* Ab = A + (long)blockIdx.y * aB;
  float*       Db = D + (long)blockIdx.y * dB;
  int half = lane >> 4, lq = lane & 15;
  int amc = imin(tm * 16 + lq, M - 1);                 // clamped A row
  int bnc = imin(tn * 16 + lq, N - 1);                 // clamped B col
  const v2f* Ap = (const v2f*)(Ab + (long)amc * K);
  const v2f* Bp = (const v2f*)(Bt + (long)bnc * K);
  int K4 = K >> 2;                                     // K-steps of 4
  v8f acc = {};
  int j = 0;
  for (; j + 2 <= K4; j += 2) {
    v2f a0 = Ap[2 * j + half],     b0 = Bp[2 * j + half];
    v2f a1 = Ap[2 * j + 2 + half], b1 = Bp[2 * j + 2 + half];
    acc = __builtin_amdgcn_wmma_f32_16x16x4_f32(false, a0, false, b0, (short)0, acc,
                                                false, false);
    acc = __builtin_amdgcn_wmma_f32_16x16x4_f32(false, a1, false, b1, (short)0, acc,
                                                false, false);
  }
  for (; j < K4; ++j) {
    v2f a0 = Ap[2 * j + half], b0 = Bp[2 * j + half];
    acc = __builtin_amdgcn_wmma_f32_16x16x4_f32(false, a0, false, b0, (short)0, acc,
                                                false, false);
  }
  for (int i = 0; i < 8; ++i) {
    int row = tm * 16 + half * 8 + i;
    int col = tn * 16 + lq;
    if (row < M && col < N)
      Db[(long)row * N + col] = acc[i] + (bias ? bias[col] : 0.f);
  }
}

// ---------------------------------------------------------------------------
// imp[b,q] = max_{t,s} q[b,q,:].k[b,t,s,:]   (positive scale dropped: monotone)
// Q tile fragments are loop-invariant -> preloaded into 16 v2f registers.
// Inner loop: one v2f load + one WMMA. Columns >= S are clamped loads and
// masked out of the max. Cross-lane row max via shfl_xor in 16-lane halves.
// ---------------------------------------------------------------------------
__global__ void imp_wmma_kernel(const float* __restrict__ Q, const float* __restrict__ Km,
                                float* __restrict__ imp, int Bn, int Tn, int Nq, int S) {
  int wave = threadIdx.x >> 5;
  int lane = threadIdx.x & 31;
  int ntm = Nq >> 4;
  int wid = blockIdx.x * (blockDim.x >> 5) + wave;
  if (wid >= Bn * ntm) return;                         // wave-uniform
  int b = wid / ntm, tm = wid % ntm;
  int half = lane >> 4, lq = lane & 15;
  const v2f* Qp = (const v2f*)(Q + ((long)b * Nq + tm * 16 + lq) * C0);
  v2f qf[C0 / 4];
  for (int j = 0; j < C0 / 4; ++j) qf[j] = Qp[2 * j + half];
  const float NEG = -3.4028235e38f;
  v8f rmax;
  for (int i = 0; i < 8; ++i) rmax[i] = NEG;
  int nst = (S + 15) >> 4;
  for (int t = 0; t < Tn; ++t) {
    const float* Kb = Km + (long)(b * Tn + t) * (long)S * C0;
    for (int st = 0; st < nst; ++st) {
      int sn = st * 16 + lq;
      const v2f* Kp = (const v2f*)(Kb + (long)imin(sn, S - 1) * C0);
      v8f acc = {};
      for (int j = 0; j < C0 / 4; ++j)
        acc = __builtin_amdgcn_wmma_f32_16x16x4_f32(false, qf[j], false,
                                                    Kp[2 * j + half], (short)0, acc,
                                                    false, false);
      if (sn < S)
        for (int i = 0; i < 8; ++i) rmax[i] = fmaxf(rmax[i], acc[i]);
    }
  }
  for (int m = 1; m < 16; m <<= 1)
    for (int i = 0; i < 8; ++i) rmax[i] = fmaxf(rmax[i], __shfl_xor(rmax[i], m, 32));
  if (lq == 0) {
    int base = b * Nq + tm * 16 + half * 8;
    for (int i = 0; i < 8; ++i) imp[base + i] = rmax[i];
  }
}

// ---------------------------------------------------------------------------
// adaptive_avg_pool2d (torch semantics, integer bin edges)
// ---------------------------------------------------------------------------
__global__ void pool_kernel(const float* __restrict__ in, float* __restrict__ tmp,
                            int BT, int C, int H, int oh) {
  int i = blockIdx.x * blockDim.x + threadIdx.x;
  int tot = BT * C * oh * oh;
  if (i >= tot) return;
  int ox = i % oh, oy = (i / oh) % oh, c = (i / (oh * oh)) % C, bt = i / (oh * oh * C);
  int h0 = (oy * H) / oh, h1 = ((oy + 1) * H + oh - 1) / oh;
  int w0 = (ox * H) / oh, w1 = ((ox + 1) * H + oh - 1) / oh;
  const float* p = in + ((long)bt * C + c) * H * H;
  float s = 0.f;
  for (int y = h0; y < h1; ++y)
    for (int x = w0; x < w1; ++x) s += p[y * H + x];
  tmp[i] = s / (float)((h1 - h0) * (w1 - w0));
}

// pool + depthwise-3x3(pool) + bias, write transposed into [BT, Stot, C]
__global__ void dwconv_t_kernel(const float* __restrict__ tmp, const float* __restrict__ w,
                                const float* __restrict__ bias, float* __restrict__ pooled,
                                int BT, int C, int oh, int Stot, int soff) {
  int i = blockIdx.x * blockDim.x + threadIdx.x;
  int tot = BT * C * oh * oh;
  if (i >= tot) return;
  int ox = i % oh, oy = (i / oh) % oh, c = (i / (oh * oh)) % C, bt = i / (oh * oh * C);
  const float* t = tmp + ((long)bt * C + c) * oh * oh;
  float conv = 0.f;
  for (int ky = 0; ky < 3; ++ky)
    for (int kx = 0; kx < 3; ++kx) {
      int yy = oy + ky - 1, xx = ox + kx - 1;
      if (yy >= 0 && yy < oh && xx >= 0 && xx < oh)
        conv += w[c * 9 + ky * 3 + kx] * t[yy * oh + xx];
    }
  float val = t[oy * oh + ox] + conv + bias[c];
  pooled[((long)bt * Stot + soff + oy * oh + ox) * C + c] = val;
}

// LayerNorm over last dim C, in place, one block per row
__global__ void ln_kernel(float* __restrict__ x, const float* __restrict__ g,
                          const float* __restrict__ b, int C) {
  long row = blockIdx.x;
  float* p = x + row * C;
  __shared__ float r1[128], r2[128];
  int tid = threadIdx.x;
  float s = 0.f, s2 = 0.f;
  for (int c = tid; c < C; c += 128) { float v = p[c]; s += v; s2 += v * v; }
  r1[tid] = s; r2[tid] = s2; __syncthreads();
  for (int o = 64; o > 0; o >>= 1) {
    if (tid < o) { r1[tid] += r1[tid + o]; r2[tid] += r2[tid + o]; }
    __syncthreads();
  }
  float m = r1[0] / (float)C;
  float var = r2[0] / (float)C - m * m;
  float r = rsqrtf(var + 1e-5f);
  for (int c = tid; c < C; c += 128) p[c] = (p[c] - m) * r * g[c] + b[c];
}

// ---------------------------------------------------------------------------
// Exact top-Nk selection as a set: binary search threshold on ordered-float
// bits, then deterministic tie fill (ascending index). One block per batch.
// ---------------------------------------------------------------------------
__global__ void select_kernel(const float* __restrict__ imp, int* __restrict__ sel,
                              int Nq, int Nk) {
  __shared__ unsigned keys[NQ0];
  __shared__ unsigned red[256];
  int b = blockIdx.x, tid = threadIdx.x;
  for (int q = tid; q < Nq; q += 256) {
    unsigned u = __float_as_uint(imp[b * Nq + q]);
    keys[q] = (u & 0x80000000u) ? ~u : (u | 0x80000000u);
  }
  __syncthreads();
  unsigned thr = 0;
  for (int bit = 31; bit >= 0; --bit) {
    unsigned cand = thr | (1u << bit);
    unsigned cnt = 0;
    for (int q = tid; q < Nq; q += 256) cnt += (keys[q] >= cand) ? 1u : 0u;
    red[tid] = cnt; __syncthreads();
    for (int o = 128; o > 0; o >>= 1) { if (tid < o) red[tid] += red[tid + o]; __syncthreads(); }
    unsigned total = red[0]; __syncthreads();
    if (total >= (unsigned)Nk) thr = cand;
  }
  unsigned cnt = 0;
  for (int q = tid; q < Nq; q += 256) cnt += (keys[q] > thr) ? 1u : 0u;
  red[tid] = cnt; __syncthreads();
  for (int o = 128; o > 0; o >>= 1) { if (tid < o) red[tid] += red[tid + o]; __syncthreads(); }
  int gt = (int)red[0];
  __syncthreads();
  if (tid == 0) {
    int extra = Nk - gt;
    for (int q = 0; q < Nq; ++q) {
      int s;
      if (keys[q] > thr) s = 1;
      else if (keys[q] == thr && extra > 0) { s = 1; --extra; }
      else s = 0;
      sel[b * Nq + q] = s;
    }
  }
}

// ---------------------------------------------------------------------------
// Selected rows only: a = scale*(q.k^T); softmax over S; out = sum p*v / T.
// One block per (b,q). Unselected rows write zeros (d_out is poisoned).
// float4-vectorized q.k dot products.
// ---------------------------------------------------------------------------
__global__ void pv_kernel(const float* __restrict__ Q, const float* __restrict__ Km,
                          const float* __restrict__ Vm, const int* __restrict__ sel,
                          float* __restrict__ out, int Bn, int Tn, int Nq, float scale) {
  int b = blockIdx.x / Nq, qi = blockIdx.x % Nq, tid = threadIdx.x;
  __shared__ float abuf[S0TOT];
  __shared__ float qrow[C0];
  __shared__ float red[256];
  long obase = ((long)b * Nq + qi) * C0;
  if (!sel[b * Nq + qi]) {
    for (int c = tid; c < C0; c += 256) out[obase + c] = 0.f;
    return;
  }
  for (int c = tid; c < C0; c += 256) qrow[c] = Q[obase + c];
  float acc = 0.f;
  for (int t = 0; t < Tn; ++t) {
    long kb = (long)(b * Tn + t) * (long)S0TOT * C0;
    __syncthreads();
    const float4* q4 = (const float4*)qrow;
    for (int s = tid; s < S0TOT; s += 256) {
      const float4* k4 = (const float4*)(Km + kb + (long)s * C0);
      float d = 0.f;
      for (int c4 = 0; c4 < C0 / 4; ++c4) {
        float4 kv = k4[c4], qv = q4[c4];
        d += kv.x * qv.x + kv.y * qv.y + kv.z * qv.z + kv.w * qv.w;
      }
      abuf[s] = d * scale;
    }
    __syncthreads();
    float lm = -3.4028235e38f;
    for (int s = tid; s < S0TOT; s += 256) lm = fmaxf(lm, abuf[s]);
    red[tid] = lm; __syncthreads();
    for (int o = 128; o > 0; o >>= 1) { if (tid < o) red[tid] = fmaxf(red[tid], red[tid + o]); __syncthreads(); }
    float m = red[0]; __syncthreads();
    float ls = 0.f;
    for (int s = tid; s < S0TOT; s += 256) { float e = __expf(abuf[s] - m); abuf[s] = e; ls += e; }
    red[tid] = ls; __syncthreads();
    for (int o = 128; o > 0; o >>= 1) { if (tid < o) red[tid] += red[tid + o]; __syncthreads(); }
    float Z = red[0]; __syncthreads();
    if (tid < C0) {
      const float* vb = Vm + kb;
      float s2 = 0.f;
      for (int s = 0; s < S0TOT; ++s) s2 += abuf[s] * vb[(long)s * C0 + tid];
      acc += s2 / Z;
    }
  }
  if (tid < C0) out[obase + tid] = acc / (float)Tn;
}

// mean of k over (t,s): ksum[b,c]
__global__ void kmean_kernel(const float* __restrict__ Km, float* __restrict__ ksum,
                             int Bn, int Tn, int S, int C) {
  int i = blockIdx.x * blockDim.x + threadIdx.x;
  if (i >= Bn * C) return;
  int b = i / C, c = i % C;
  float s = 0.f;
  for (int t = 0; t < Tn; ++t)
    for (int ss = 0; ss < S; ++ss) s += Km[((long)(b * Tn + t) * S + ss) * C + c];
  ksum[i] = s / (float)(Tn * S);
}

// corr[b,n] = scale * q[b,n,:].ksum[b,:]   (== atten.mean over t,s)
__global__ void corr_kernel(const float* __restrict__ Q, const float* __restrict__ ksum,
                            float* __restrict__ corr, int Nq, int C, float scale, int tot) {
  int i = blockIdx.x * blockDim.x + threadIdx.x;
  if (i >= tot) return;
  int b = i / Nq, n = i % Nq;
  const float4* q4 = (const float4*)(Q + ((long)b * Nq + n) * C);
  const float4* k4 = (const float4*)(ksum + b * C);
  float d = 0.f;
  for (int c4 = 0; c4 < C / 4; ++c4) {
    float4 qv = q4[c4], kv = k4[c4];
    d += qv.x * kv.x + qv.y * kv.y + qv.z * kv.z + qv.w * kv.w;
  }
  corr[i] = d * scale;
}

// conv(1ch, kxk, pad) + bias -> per-sample mean/var norm * g + bt -> relu
__global__ void cgn_kernel(const float* __restrict__ in, const float* __restrict__ add,
                           const float* __restrict__ w, const float* __restrict__ bs,
                           const float* __restrict__ g, const float* __restrict__ bt,
                           float* __restrict__ out, int H, int ksz) {
  int b = blockIdx.x, tid = threadIdx.x;
  int HW = H * H;
  int y = tid / H, x = tid % H, pad = ksz / 2;
  __shared__ float r1[1024], r2[1024];
  const float* ib = in + (long)b * HW;
  const float* ab = add ? add + (long)b * HW : (const float*)0;
  float val = bs[0];
  for (int ky = 0; ky < ksz; ++ky)
    for (int kx = 0; kx < ksz; ++kx) {
      int yy = y + ky - pad, xx = x + kx - pad;
      if (yy >= 0 && yy < H && xx >= 0 && xx < H) {
        float s = ib[yy * H + xx] + (ab ? ab[yy * H + xx] : 0.f);
        val += w[ky * ksz + kx] * s;
      }
    }
  r1[tid] = val; r2[tid] = val * val; __syncthreads();
  for (int o = HW >> 1; o > 0; o >>= 1) {
    if (tid < o) { r1[tid] += r1[tid + o]; r2[tid] += r2[tid + o]; }
    __syncthreads();
  }
  float m = r1[0] / (float)HW;
  float var = r2[0] / (float)HW - m * m;
  float o = (val - m) * rsqrtf(var + 1e-5f) * g[0] + bt[0];
  out[(long)b * HW + tid] = fmaxf(o, 0.f);
}

// bilinear resize, align_corners=True (square, ih>1, oh>1)
__global__ void resize_kernel(const float* __restrict__ in, float* __restrict__ out,
                              int Bn, int ih, int oh) {
  int i = blockIdx.x * blockDim.x + threadIdx.x;
  if (i >= Bn * oh * oh) return;
  int x = i % oh, y = (i / oh) % oh, b = i / (oh * oh);
  float fy = (float)y * (float)(ih - 1) / (float)(oh - 1);
  float fx = (float)x * (float)(ih - 1) / (float)(oh - 1);
  int y0 = (int)fy, x0 = (int)fx;
  int y1 = imin(y0 + 1, ih - 1), x1 = imin(x0 + 1, ih - 1);
  float ty = fy - (float)y0, tx = fx - (float)x0;
  const float* p = in + (long)b * ih * ih;
  float a = p[y0 * ih + x0] * (1.f - tx) + p[y0 * ih + x1] * tx;
  float c = p[y1 * ih + x0] * (1.f - tx) + p[y1 * ih + x1] * tx;
  out[i] = a * (1.f - ty) + c * ty;
}

// out[b,q,c] *= (1 + h64[b,q])
__global__ void final_kernel(float* __restrict__ out, const float* __restrict__ h64,
                             int Nq, int C, long total) {
  long i = (long)blockIdx.x * blockDim.x + threadIdx.x;
  if (i >= total) return;
  long bq = i / C;
  int b = (int)(bq / Nq), q = (int)(bq % Nq);
  out[i] *= (1.f + h64[(long)b * Nq + q]);
}

// ---------------------------------------------------------------------------
extern "C" void kernel_launch(void* const* d_in, const int* in_sizes, int n_in,
                              void* d_out, int out_size, void* d_ws, size_t ws_size,
                              hipStream_t stream) {
  (void)in_sizes; (void)n_in; (void)out_size; (void)ws_size;
  const int Bn = 2, Tn = 3, BT = 6;
  static const int DIMS[4] = {64, 128, 320, 512};
  static const int HWS[4]  = {64, 32, 16, 8};
  static const int OHS[4][4] = {{32, 16, 11, 8}, {16, 8, 5, 4}, {8, 4, 3, 2}, {4, 2, 1, 1}};
  static const int STOT[4] = {1465, 361, 93, 22};

  auto P = [&](int i) { return (const float*)d_in[i]; };
  float* ws = (float*)d_ws;
  size_t off = 0;
  auto alloc = [&](size_t n) { size_t r = off; off += (n + 63) & ~(size_t)63; return r; };
  float* pooled = ws + alloc((size_t)BT * 1465 * 64);   // reused per scale
  float* tmpP   = ws + alloc((size_t)BT * 64 * 32 * 32);
  float* kbuf   = ws + alloc((size_t)BT * 1465 * 64);   // reused per scale
  float* vbuf   = ws + alloc((size_t)BT * 1465 * 64);
  float* qbuf   = ws + alloc((size_t)Bn * NQ0 * 64);    // reused per scale
  float* wtbuf  = ws + alloc((size_t)512 * 512);        // transposed weight
  float* xtbuf  = ws + alloc((size_t)Bn * NQ0 * 64);    // transposed x tokens
  float* impb   = ws + alloc((size_t)Bn * NQ0);
  int*   selb   = (int*)(ws + alloc((size_t)Bn * NQ0));
  float* ksumB  = ws + alloc((size_t)Bn * 512);
  float* corr1  = ws + alloc((size_t)Bn * 1024);
  float* corr2  = ws + alloc((size_t)Bn * 256);
  float* corr3  = ws + alloc((size_t)Bn * 64);
  float* hs4  = ws + alloc((size_t)Bn * 64);
  float* hs3  = ws + alloc((size_t)Bn * 256);
  float* hs2  = ws + alloc((size_t)Bn * 1024);
  float* ht16 = ws + alloc((size_t)Bn * 256);
  float* hm43 = ws + alloc((size_t)Bn * 256);
  float* ht32 = ws + alloc((size_t)Bn * 1024);
  float* hhot = ws + alloc((size_t)Bn * 1024);
  float* hh64 = ws + alloc((size_t)Bn * NQ0);
  float* outF = (float*)d_out;

  auto trans = [&](const float* in, float* out, int R, int Cc, int nb, long ib, long ob) {
    dim3 g(((R * Cc) + 255) / 256, nb);
    transpose_kernel<<<g, 256, 0, stream>>>(in, out, R, Cc, ib, ob);
  };
  // D = A @ W + bias, A row-major [M,K]; W [K,N] transposed into wtbuf first.
  auto gemm = [&](const float* A, long aB, int nb, const float* W, const float* bias,
                  float* D, long dB, int M, int N, int K) {
    trans(W, wtbuf, K, N, 1, 0, 0);
    int tiles = ((M + 15) / 16) * ((N + 15) / 16);
    dim3 grid((tiles + 3) / 4, nb);
    gemm_wmma_kernel<<<grid, 128, 0, stream>>>(A, aB, wtbuf, bias, D, dB, M, N, K);
  };

  auto do_pooling = [&](int scale, const float* r, const float* dwW, const float* dwB,
                        const float* lng, const float* lnb) {
    int C = DIMS[scale], hw = HWS[scale], S = STOT[scale];
    int soff = 0;
    for (int ri = 0; ri < 4; ++ri) {
      int oh = OHS[scale][ri];
      int n1 = BT * C * oh * oh;
      pool_kernel<<<(n1 + 255) / 256, 256, 0, stream>>>(r, tmpP, BT, C, hw, oh);
      dwconv_t_kernel<<<(n1 + 255) / 256, 256, 0, stream>>>(
          tmpP, dwW + (size_t)ri * C * 9, dwB + (size_t)ri * C, pooled, BT, C, oh, S, soff);
      soff += oh * oh;
    }
    ln_kernel<<<BT * S, 128, 0, stream>>>(pooled, lng, lnb, C);
  };

  // ------------------------- scale 0 (full attention) ----------------------
  {
    const int sc = 0, pb = 28;
    const float *Wk = P(pb + 0), *Wq = P(pb + 1), *Wv = P(pb + 2);
    const float *bk = P(pb + 3), *bq = P(pb + 4), *bv = P(pb + 5);
    const float *dwB = P(pb + 6), *dwW = P(pb + 7), *lnb = P(pb + 8), *lng = P(pb + 9);
    int C = DIMS[sc], Nq = NQ0, S = STOT[sc];
    do_pooling(sc, P(1), dwW, dwB, lng, lnb);
    gemm(pooled, 0, 1, Wk, bk, kbuf, 0, BT * S, C, C);
    gemm(pooled, 0, 1, Wv, bv, vbuf, 0, BT * S, C, C);
    // q: x0 [B,1,C,64,64] -> token-major [B,4096,C] via transpose, then GEMM
    trans(P(0), xtbuf, C, Nq, Bn, (long)C * Nq, (long)Nq * C);
    gemm(xtbuf, (long)Nq * C, Bn, Wq, bq, qbuf, (long)Nq * C, Nq, C, C);
    int waves = Bn * (Nq / 16);
    imp_wmma_kernel<<<(waves + 3) / 4, 128, 0, stream>>>(qbuf, kbuf, impb, Bn, Tn, Nq, S);
    select_kernel<<<Bn, 256, 0, stream>>>(impb, selb, Nq, Nq / 2);
    float scl = 1.f / sqrtf((float)C);
    pv_kernel<<<Bn * Nq, 256, 0, stream>>>(qbuf, kbuf, vbuf, selb, outF, Bn, Tn, Nq, scl);
  }

  // --------------------- coarse scales: mean-k fusion ----------------------
  float* corrDst[4] = {0, corr1, corr2, corr3};
  for (int sc = 1; sc < 4; ++sc) {
    int pb = 28 + 10 * sc;
    const float *Wk = P(pb + 0), *Wq = P(pb + 1);
    const float *bk = P(pb + 3), *bq = P(pb + 4);
    const float *dwB = P(pb + 6), *dwW = P(pb + 7), *lnb = P(pb + 8), *lng = P(pb + 9);
    int C = DIMS[sc], hw = HWS[sc], S = STOT[sc], Nq = hw * hw;
    do_pooling(sc, P(2 * sc + 1), dwW, dwB, lng, lnb);
    gemm(pooled, 0, 1, Wk, bk, kbuf, 0, BT * S, C, C);
    kmean_kernel<<<(Bn * C + 255) / 256, 256, 0, stream>>>(kbuf, ksumB, Bn, Tn, S, C);
    trans(P(2 * sc), xtbuf, C, Nq, Bn, (long)C * Nq, (long)Nq * C);
    gemm(xtbuf, (long)Nq * C, Bn, Wq, bq, qbuf, (long)Nq * C, Nq, C, C);
    float scl = 1.f / sqrtf((float)C);
    corr_kernel<<<(Bn * Nq + 255) / 256, 256, 0, stream>>>(qbuf, ksumB, corrDst[sc], Nq, C,
                                                           scl, Bn * Nq);
  }

  // ------------------------------- HPN -------------------------------------
  // hpn leaves: l2:8-11  l3:12-15  l32:16-19  l4:20-23  l43:24-27  ({b,bt,g,w})
  cgn_kernel<<<Bn, 8 * 8, 0, stream>>>(corr3, (const float*)0, P(23), P(20), P(22), P(21),
                                       hs4, 8, 3);                         // l4, k=3
  cgn_kernel<<<Bn, 16 * 16, 0, stream>>>(corr2, (const float*)0, P(15), P(12), P(14), P(13),
                                         hs3, 16, 5);                      // l3, k=5
  cgn_kernel<<<Bn, 32 * 32, 0, stream>>>(corr1, (const float*)0, P(11), P(8), P(10), P(9),
                                         hs2, 32, 5);                      // l2, k=5
  resize_kernel<<<(Bn * 256 + 255) / 256, 256, 0, stream>>>(hs4, ht16, Bn, 8, 16);
  cgn_kernel<<<Bn, 16 * 16, 0, stream>>>(ht16, hs3, P(27), P(24), P(26), P(25),
                                         hm43, 16, 3);                     // l43, k=3
  resize_kernel<<<(Bn * 1024 + 255) / 256, 256, 0, stream>>>(hm43, ht32, Bn, 16, 32);
  cgn_kernel<<<Bn, 32 * 32, 0, stream>>>(ht32, hs2, P(19), P(16), P(18), P(17),
                                         hhot, 32, 3);                     // l32, k=3
  resize_kernel<<<(Bn * NQ0 + 255) / 256, 256, 0, stream>>>(hhot, hh64, Bn, 32, 64);

  long total = (long)Bn * NQ0 * 64;
  final_kernel<<<(int)((total + 255) / 256), 256, 0, stream>>>(outF, hh64, NQ0, 64, total);
}